// Embedding_63694364999796
// MI455X (gfx1250) — compile-verified
//
#include <hip/hip_runtime.h>
#include <hip/hip_bf16.h>

// Problem constants (match reference)
#define NNODES 100000
#define SUBD   128
#define BASED  128
#define FINALD 128
#define LDIM   4
#define DIMA   64
#define LAMBF  0.5f
#define BDIM   4096
#define KNEI   32

typedef __attribute__((ext_vector_type(2))) float v2f;
typedef __attribute__((ext_vector_type(8))) float v8f;

// ---------------------------------------------------------------------------
// Kernel 1: gather neighbor rows + mean over K  ->  nte[(b*L+l)*128 + s]
// One wave32 per (b,l) row; each lane owns one float4 (4 of 128 floats).
// This is the bandwidth-dominant stage (~256MB of 512B-row gathers).
// ---------------------------------------------------------------------------
__global__ void gather_mean_kernel(const int* __restrict__ node_neigh,
                                   const float* __restrict__ sub_embedding,
                                   float* __restrict__ nte) {
    const int wave = (blockIdx.x * blockDim.x + threadIdx.x) >> 5;   // = b*L + l
    const int lane = threadIdx.x & 31;
    if (wave >= BDIM * LDIM) return;
    const int b = wave >> 2;          // L = 4
    const int l = wave & 3;

    const int* nn = node_neigh + (size_t)b * (LDIM * KNEI) + (size_t)l * KNEI;

    float4 acc = make_float4(0.f, 0.f, 0.f, 0.f);
    int n = nn[0];
#pragma unroll 4
    for (int k = 0; k < KNEI; ++k) {
        const float4* row =
            (const float4*)(sub_embedding + ((size_t)n * LDIM + l) * SUBD);
        const int n_next = (k + 1 < KNEI) ? nn[k + 1] : n;
        // CDNA5 global_prefetch_b8 of the next neighbor row
        __builtin_prefetch(sub_embedding + ((size_t)n_next * LDIM + l) * SUBD + lane * 4, 0, 0);
        float4 v = row[lane];
        acc.x += v.x; acc.y += v.y; acc.z += v.z; acc.w += v.w;
        n = n_next;
    }
    const float inv = 1.0f / (float)KNEI;
    float4 o = make_float4(acc.x * inv, acc.y * inv, acc.z * inv, acc.w * inv);
    ((float4*)(nte + (size_t)wave * SUBD))[lane] = o;
}

// ---------------------------------------------------------------------------
// Kernel 2/4: D = relu(A @ Bw), A: M x Kdim (row-major), Bw: Kdim x 128, D: M x 128
// One wave32 per 16x16 output tile; Kdim/4 chained V_WMMA_F32_16X16X4_F32.
// A-frag (16x4 f32): lane m = lane&15; VGPR0 holds K= (lane<16?0:2), VGPR1 K+1.
// B-frag (4x16 f32): lane n = lane&15; same half-wave K split.
// D (16x16 f32, 8 VGPRs): row = r + (lane<16?0:8), col = lane&15.
// ---------------------------------------------------------------------------
__global__ void wmma_gemm_relu_kernel(const float* __restrict__ A,
                                      const float* __restrict__ Bw,
                                      float* __restrict__ D,
                                      int Kdim, int Mtiles) {
    const int gwave = (blockIdx.x * blockDim.x + threadIdx.x) >> 5;
    const int lane  = threadIdx.x & 31;
    const int tn = gwave & 7;        // 128 / 16 = 8 N-tiles
    const int tm = gwave >> 3;
    if (tm >= Mtiles) return;        // wave-uniform: EXEC stays all-1s for WMMA

    const int half = lane >> 4;      // 0: K%4 in {0,1}, 1: K%4 in {2,3}
    const int idx  = lane & 15;

    const float* Arow = A  + (size_t)(tm * 16 + idx) * Kdim;
    const float* Bcol = Bw + (size_t)(tn * 16 + idx);

    v8f acc = {};
    const int kblocks = Kdim >> 2;
    for (int kb = 0; kb < kblocks; ++kb) {
        const int k0 = (kb << 2) + (half << 1);
        v2f a, b;
        a.x = Arow[k0];
        a.y = Arow[k0 + 1];
        b.x = Bcol[(size_t)k0 * FINALD];
        b.y = Bcol[(size_t)(k0 + 1) * FINALD];
        acc = __builtin_amdgcn_wmma_f32_16x16x4_f32(
            /*neg_a=*/false, a, /*neg_b=*/false, b,
            /*c_mod=*/(short)0, acc, /*reuse_a=*/false, /*reuse_b=*/false);
    }

    const int rowbase = tm * 16 + half * 8;
    float* Dp = D + (size_t)rowbase * FINALD + (size_t)(tn * 16 + idx);
#pragma unroll
    for (int r = 0; r < 8; ++r) {
        float v = acc[r];
        Dp[(size_t)r * FINALD] = v > 0.0f ? v : 0.0f;
    }
}

// ---------------------------------------------------------------------------
// Kernel 3: attention scores + softmax(L=4) + integrate + build fused rows.
// One wave32 per batch element b. Lanes cover the 64 attention dims (2 each).
// fused[b, 0:128]   = base_embeddings[inputs[b]]
// fused[b, 128:256] = 0.5 * sum_l att[b,l] * h[b,l,:]
// ---------------------------------------------------------------------------
__global__ void attn_fuse_kernel(const float* __restrict__ Hm,       // (B*L) x 128
                                 const float* __restrict__ attn_w,   // 128 x 64
                                 const float* __restrict__ attn_a,   // 64
                                 const float* __restrict__ attn_b,   // 64
                                 const int*   __restrict__ inputs,
                                 const float* __restrict__ base_emb, // NNODES x 128
                                 float* __restrict__ fused) {        // B x 256
    const int wave = (blockIdx.x * blockDim.x + threadIdx.x) >> 5;   // = b
    const int lane = threadIdx.x & 31;
    if (wave >= BDIM) return;
    const int b = wave;

    const float* hb = Hm + (size_t)b * LDIM * SUBD;

    float sc[LDIM];
#pragma unroll
    for (int l = 0; l < LDIM; ++l) {
        float acc0 = attn_b[lane];
        float acc1 = attn_b[lane + 32];
        const float* hl = hb + (size_t)l * SUBD;
        for (int s = 0; s < SUBD; ++s) {
            const float hv = hl[s];
            acc0 += hv * attn_w[(size_t)s * DIMA + lane];
            acc1 += hv * attn_w[(size_t)s * DIMA + lane + 32];
        }
        float t = tanhf(acc0) * attn_a[lane] + tanhf(acc1) * attn_a[lane + 32];
#pragma unroll
        for (int off = 16; off > 0; off >>= 1)
            t += __shfl_xor(t, off, 32);
        sc[l] = t;   // full dot over 64 dims, replicated to all lanes
    }

    // softmax over L=4 (uniform across the wave)
    float m = fmaxf(fmaxf(sc[0], sc[1]), fmaxf(sc[2], sc[3]));
    float e0 = expf(sc[0] - m), e1 = expf(sc[1] - m);
    float e2 = expf(sc[2] - m), e3 = expf(sc[3] - m);
    const float inv = 1.0f / (e0 + e1 + e2 + e3);
    const float att0 = e0 * inv, att1 = e1 * inv, att2 = e2 * inv, att3 = e3 * inv;

    float* fb = fused + (size_t)b * (BASED + SUBD);

    // base half (gather one row of base embeddings)
    const float4 base =
        ((const float4*)(base_emb + (size_t)inputs[b] * BASED))[lane];
    ((float4*)fb)[lane] = base;

    // integrated half: lane owns 4 of 128 elements
    float4 h0 = ((const float4*)(hb + 0 * SUBD))[lane];
    float4 h1 = ((const float4*)(hb + 1 * SUBD))[lane];
    float4 h2 = ((const float4*)(hb + 2 * SUBD))[lane];
    float4 h3 = ((const float4*)(hb + 3 * SUBD))[lane];
    float4 g;
    g.x = LAMBF * (att0 * h0.x + att1 * h1.x + att2 * h2.x + att3 * h3.x);
    g.y = LAMBF * (att0 * h0.y + att1 * h1.y + att2 * h2.y + att3 * h3.y);
    g.z = LAMBF * (att0 * h0.z + att1 * h1.z + att2 * h2.z + att3 * h3.z);
    g.w = LAMBF * (att0 * h0.w + att1 * h1.w + att2 * h2.w + att3 * h3.w);
    ((float4*)(fb + BASED))[lane] = g;
}

// ---------------------------------------------------------------------------
extern "C" void kernel_launch(void* const* d_in, const int* in_sizes, int n_in,
                              void* d_out, int out_size, void* d_ws, size_t ws_size,
                              hipStream_t stream) {
    const int*   inputs     = (const int*)d_in[0];
    const int*   node_neigh = (const int*)d_in[1];
    const float* base_emb   = (const float*)d_in[2];
    const float* sub_emb    = (const float*)d_in[3];
    const float* w1         = (const float*)d_in[4];
    const float* attn_w     = (const float*)d_in[5];
    const float* attn_a     = (const float*)d_in[6];
    const float* attn_b     = (const float*)d_in[7];
    const float* w3         = (const float*)d_in[8];
    float* out = (float*)d_out;

    // Workspace layout (floats): nte (16384*128) | h (16384*128) | fused (4096*256)
    float* nte   = (float*)d_ws;
    float* h     = nte + (size_t)BDIM * LDIM * SUBD;
    float* fused = h   + (size_t)BDIM * LDIM * SUBD;

    // 1) gather + mean: 16384 waves, 8 waves/block
    gather_mean_kernel<<<(BDIM * LDIM) / 8, 256, 0, stream>>>(node_neigh, sub_emb, nte);

    // 2) h = relu(nte @ w1): M=16384, K=128 -> 8192 tiles -> 1024 blocks
    wmma_gemm_relu_kernel<<<((BDIM * LDIM / 16) * 8) / 8, 256, 0, stream>>>(
        nte, w1, h, SUBD, BDIM * LDIM / 16);

    // 3) attention + softmax + fused rows: 4096 waves
    attn_fuse_kernel<<<BDIM / 8, 256, 0, stream>>>(
        h, attn_w, attn_a, attn_b, inputs, base_emb, fused);

    // 4) out = relu(fused @ w3): M=4096, K=256 -> 2048 tiles -> 256 blocks
    wmma_gemm_relu_kernel<<<((BDIM / 16) * 8) / 8, 256, 0, stream>>>(
        fused, w3, out, BASED + SUBD, BDIM / 16);
}